// MaskedGatedFusion_88734024335569
// MI455X (gfx1250) — compile-verified
//
#include <hip/hip_runtime.h>

typedef __attribute__((ext_vector_type(2))) float v2f;
typedef __attribute__((ext_vector_type(4))) float v4f;
typedef __attribute__((ext_vector_type(8))) float v8f;

#define T_N 1024
#define S_N 128
#define H_N 512
#define LN_EPS 1e-5f

// ---------------------------------------------------------------------------
// GEMM: C[M,N] = A[M,K] @ W^T  where B[k][n] = W[n*ldw + k]  (wt/ws slices of
// gate_w). One wave32 per 16x16 tile, V_WMMA_F32_16X16X4_F32 over K in steps
// of 4. fp32 WMMA keeps reference precision; GEMM cost is negligible vs the
// 268MB gates store, so no reason to downcast.
// ---------------------------------------------------------------------------
__global__ __launch_bounds__(32) void wmma_gemm_f32(
    const float* __restrict__ A,   // M x K row-major
    const float* __restrict__ W,   // B[k][n] = W[n*ldw + k]
    float* __restrict__ C,         // M x N row-major
    int M, int N, int K, int ldw)
{
  const int lane = threadIdx.x;            // wave32
  const int tm = blockIdx.y * 16;
  const int tn = blockIdx.x * 16;

  // A frag: lanes 0-15 -> (M=lane, K=k0+{0,1}); lanes 16-31 -> (M=lane-16, K=k0+{2,3})
  const int arow  = tm + (lane & 15);
  const int khalf = (lane >> 4) * 2;       // 0 or 2
  // B frag (4x16, K x N): lanes 0-15 -> K=k0+{0,1}, N=lane; lanes 16-31 -> K=k0+{2,3}
  const int bcol  = tn + (lane & 15);

  const float* Arow = A + (size_t)arow * K + khalf;
  const float* Wcol = W + (size_t)bcol * ldw + khalf;

  v8f c = {};
  for (int k0 = 0; k0 < K; k0 += 4) {
    v2f a = *(const v2f*)(Arow + k0);
    v2f b = *(const v2f*)(Wcol + k0);
    // (neg_a, A, neg_b, B, c_mod, C, reuse_a, reuse_b)
    c = __builtin_amdgcn_wmma_f32_16x16x4_f32(false, a, false, b,
                                              (short)0, c, false, false);
  }

  // D layout: VGPR i -> M = tm + i + 8*(lane>=16), N = tn + (lane&15)
  const int mbase = tm + (lane >> 4) * 8;
  const int ccol  = tn + (lane & 15);
#pragma unroll
  for (int i = 0; i < 8; ++i) {
    C[(size_t)(mbase + i) * N + ccol] = c[i];
  }
}

// ---------------------------------------------------------------------------
// Fused: gates = relu(gt[t]+gs[s]+b)*mask[t,s]  (268MB NT stream-out),
// enrichment[t] = sum_s gates[t,s,:]*struct[s,:]  (pointwise in h),
// enriched = layernorm(text + enrichment).
// One block per t; 128 threads x float4 covers H=512. Write-BW bound.
// ---------------------------------------------------------------------------
__global__ __launch_bounds__(128) void fused_gate_enrich_ln(
    const float* __restrict__ text,    // T x H
    const float* __restrict__ strct,   // S x H
    const int*   __restrict__ mask,    // T x S
    const float* __restrict__ gt,      // T x H
    const float* __restrict__ gs,      // S x H
    const float* __restrict__ gate_b,  // H
    const float* __restrict__ gamma,   // H
    const float* __restrict__ beta,    // H
    float* __restrict__ out_enriched,  // T x H
    float* __restrict__ out_gates)     // T x S x H
{
  const int t   = blockIdx.x;
  const int tid = threadIdx.x;     // 128
  const int h4  = tid * 4;

  __shared__ float smask[S_N];
  __shared__ float red_sum[4];
  __shared__ float red_ssq[4];

  smask[tid] = (float)mask[t * S_N + tid];   // tid == S_N count
  __syncthreads();

  const v4f gt4 = *(const v4f*)(gt + (size_t)t * H_N + h4);
  const v4f b4  = *(const v4f*)(gate_b + h4);
  v4f acc = {};

  float* gout = out_gates + (size_t)t * S_N * H_N + h4;
  for (int s = 0; s < S_N; ++s) {
    const float m   = smask[s];
    const v4f   gsv = *(const v4f*)(gs    + (size_t)s * H_N + h4);
    const v4f   sv  = *(const v4f*)(strct + (size_t)s * H_N + h4);
    v4f g = gt4 + gsv + b4;
    g.x = fmaxf(g.x, 0.f) * m;
    g.y = fmaxf(g.y, 0.f) * m;
    g.z = fmaxf(g.z, 0.f) * m;
    g.w = fmaxf(g.w, 0.f) * m;
    // streaming write-once data: non-temporal so it doesn't thrash L2
    __builtin_nontemporal_store(g, (v4f*)(gout + (size_t)s * H_N));
    acc += g * sv;
  }

  // layernorm over H of (text + enrichment)
  const v4f x = *(const v4f*)(text + (size_t)t * H_N + h4) + acc;
  float sum = x.x + x.y + x.z + x.w;
  float ssq = x.x * x.x + x.y * x.y + x.z * x.z + x.w * x.w;

  // wave32 reduce then cross-wave via LDS (4 waves)
#pragma unroll
  for (int off = 16; off > 0; off >>= 1) {
    sum += __shfl_down(sum, off, 32);
    ssq += __shfl_down(ssq, off, 32);
  }
  if ((tid & 31) == 0) {
    red_sum[tid >> 5] = sum;
    red_ssq[tid >> 5] = ssq;
  }
  __syncthreads();
  const float tsum = red_sum[0] + red_sum[1] + red_sum[2] + red_sum[3];
  const float tssq = red_ssq[0] + red_ssq[1] + red_ssq[2] + red_ssq[3];
  const float mu   = tsum * (1.0f / H_N);
  const float var  = tssq * (1.0f / H_N) - mu * mu;
  const float rstd = rsqrtf(var + LN_EPS);

  const v4f gm = *(const v4f*)(gamma + h4);
  const v4f bt = *(const v4f*)(beta + h4);
  v4f y = (x - mu) * rstd * gm + bt;
  *(v4f*)(out_enriched + (size_t)t * H_N + h4) = y;
}

extern "C" void kernel_launch(void* const* d_in, const int* in_sizes, int n_in,
                              void* d_out, int out_size, void* d_ws, size_t ws_size,
                              hipStream_t stream) {
  const float* text   = (const float*)d_in[0];   // T x H
  const float* strct  = (const float*)d_in[1];   // S x H
  const int*   mask   = (const int*)  d_in[2];   // T x S
  const float* gate_w = (const float*)d_in[3];   // H x 2H row-major
  const float* gate_b = (const float*)d_in[4];   // H
  const float* gamma  = (const float*)d_in[5];   // H
  const float* beta   = (const float*)d_in[6];   // H

  // outputs: (enriched [T,H], gates [T,S,H]) concatenated
  float* out_enriched = (float*)d_out;
  float* out_gates    = (float*)d_out + (size_t)T_N * H_N;

  // scratch: gt then gs
  float* gt = (float*)d_ws;                        // T x H  (2 MB)
  float* gs = gt + (size_t)T_N * H_N;              // S x H  (256 KB)

  const int ldw = 2 * H_N;  // gate_w row stride

  // gt = text @ wt.T : wt[n][k] = gate_w[n*ldw + k]
  {
    dim3 grid(H_N / 16, T_N / 16);   // (32, 64)
    wmma_gemm_f32<<<grid, 32, 0, stream>>>(text, gate_w, gt,
                                           T_N, H_N, H_N, ldw);
  }
  // gs = struct @ ws.T : ws[n][k] = gate_w[n*ldw + H + k]
  {
    dim3 grid(H_N / 16, S_N / 16);   // (32, 8)
    wmma_gemm_f32<<<grid, 32, 0, stream>>>(strct, gate_w + H_N, gs,
                                           S_N, H_N, H_N, ldw);
  }
  // fused gates / enrichment / layernorm
  fused_gate_enrich_ln<<<T_N, 128, 0, stream>>>(text, strct, mask, gt, gs,
                                                gate_b, gamma, beta,
                                                out_enriched, out_gates);
}